// SPOCOConsistencyLoss_41154376630493
// MI455X (gfx1250) — compile-verified
//
#include <hip/hip_runtime.h>
#include <hip/hip_bf16.h>

// ---------------------------------------------------------------------------
// SPOCO consistency loss for MI455X (gfx1250, wave32, WMMA).
//
// Shapes: B=4, C=64, H=W=384 (HW=147456), A=30 anchors (padded to 32).
// p = exp(-max(||e||^2 - 2 e.a + ||a||^2, 0) / TWO_SIGMA)
// loss = sum_{b,a} 1 - 2*sum_hw(pq*pk) / max(sum_hw(pq^2)+sum_hw(pk^2), eps)
//
// Memory-bound: 302 MB of embeddings read once (~13us @ 23.3 TB/s).
// Dot products via V_WMMA_F32_16X16X4_F32 (full fp32 precision).
// ---------------------------------------------------------------------------

typedef __attribute__((ext_vector_type(2))) float v2f;
typedef __attribute__((ext_vector_type(8))) float v8f;

#define BB   4
#define CC   64
#define HWn  147456          // 384*384
#define AA   30
#define APAD 32

// -ln(0.9) / 0.75^2
#define INV_TS 0.18730758f
#define EPSF   1e-7f

// Pixels handled per wave / per block in the main kernel
#define PIX_PER_WAVE  64
#define PIX_PER_BLOCK 512    // 8 waves * 64
#define BLOCKS_PER_B  (HWn / PIX_PER_BLOCK)   // 288
#define NBLOCKS       (BB * BLOCKS_PER_B)     // 1152

// ---------------------------------------------------------------------------
// Kernel 1: gather anchors [B, 32, 64] (padded with zeros) from emb layout
// [B, C, H*W], and per-anchor squared norms [B, 32]. One wave per (src,b,a).
// ---------------------------------------------------------------------------
__global__ __launch_bounds__(32) void spoco_gather_anchors(
    const float* __restrict__ emb_q, const float* __restrict__ emb_k,
    const int* __restrict__ anchor_idx,
    float* __restrict__ ancQ, float* __restrict__ ancK,
    float* __restrict__ asqQ, float* __restrict__ asqK) {
  int gid  = blockIdx.x;          // 0..255
  int src  = gid >> 7;            // 0 = q, 1 = k
  int b    = (gid >> 5) & 3;
  int a    = gid & 31;
  int lane = threadIdx.x;         // wave32

  const float* emb = src ? emb_k : emb_q;
  float* anc = src ? ancK : ancQ;
  float* asq = src ? asqK : asqQ;

  float v0 = 0.0f, v1 = 0.0f;
  if (a < AA) {
    int idx = anchor_idx[b * AA + a];
    size_t base = (size_t)b * CC * HWn + (size_t)idx;
    v0 = emb[base + (size_t)lane * HWn];
    v1 = emb[base + (size_t)(lane + 32) * HWn];
  }
  size_t ao = (size_t)(b * APAD + a) * CC;
  anc[ao + lane]      = v0;
  anc[ao + lane + 32] = v1;

  float s = v0 * v0 + v1 * v1;
  #pragma unroll
  for (int off = 16; off > 0; off >>= 1) s += __shfl_xor(s, off, 32);
  if (lane == 0) asq[b * APAD + a] = s;
}

// ---------------------------------------------------------------------------
// Kernel 2: main. One block = 512 pixels of one batch; one wave = 64 pixels
// (4 M-tiles of 16). Per M-tile: accumulate 16x16 dot tiles for anchors 0-15
// and 16-31, for both q and k, with V_WMMA_F32_16X16X4_F32 over 16 K-steps.
// ---------------------------------------------------------------------------
__global__ __launch_bounds__(256) void spoco_main(
    const float* __restrict__ emb_q, const float* __restrict__ emb_k,
    const float* __restrict__ ancQ, const float* __restrict__ ancK,
    const float* __restrict__ asqQ, const float* __restrict__ asqK,
    float* __restrict__ gNum, float* __restrict__ gDq, float* __restrict__ gDk) {
  __shared__ float sAncQ[APAD * CC];   // 8 KB
  __shared__ float sAncK[APAD * CC];   // 8 KB
  __shared__ float sNum[APAD], sDq[APAD], sDk[APAD];

  int b     = blockIdx.x / BLOCKS_PER_B;
  int p0blk = (blockIdx.x % BLOCKS_PER_B) * PIX_PER_BLOCK;
  int tid   = threadIdx.x;

  // Stage anchors of this batch into LDS (coalesced).
  for (int i = tid; i < APAD * CC; i += 256) {
    sAncQ[i] = ancQ[(size_t)b * APAD * CC + i];
    sAncK[i] = ancK[(size_t)b * APAD * CC + i];
  }
  if (tid < APAD) { sNum[tid] = 0.0f; sDq[tid] = 0.0f; sDk[tid] = 0.0f; }
  __syncthreads();

  int wave = tid >> 5;
  int lane = tid & 31;
  int l15  = lane & 15;
  int kb   = (lane >> 4) << 1;        // 0 for lanes 0-15, 2 for lanes 16-31
  bool hi  = (lane & 16) != 0;
  int p0w  = p0blk + wave * PIX_PER_WAVE;

  const float* qb = emb_q + (size_t)b * CC * HWn;
  const float* kbp = emb_k + (size_t)b * CC * HWn;

  float aq0 = asqQ[b * APAD + l15], aq1 = asqQ[b * APAD + 16 + l15];
  float ak0 = asqK[b * APAD + l15], ak1 = asqK[b * APAD + 16 + l15];

  float num0 = 0.f, num1 = 0.f, dq0 = 0.f, dq1 = 0.f, dk0 = 0.f, dk1 = 0.f;

  for (int t = 0; t < 4; ++t) {
    int m0  = p0w + t * 16;
    int pix = m0 + l15;
    v8f cq0 = {}, cq1 = {}, ck0 = {}, ck1 = {};
    float esq = 0.f, esk = 0.f;

    #pragma unroll 4
    for (int k0 = 0; k0 < CC; k0 += 4) {
      int c0 = k0 + kb;
      v2f aqf, akf, bq0, bq1, bk0, bk1;
      // A fragment: 16 pixels x 4 channels (ISA 16x4 fp32 layout)
      aqf.x = qb[(size_t)c0 * HWn + pix];
      aqf.y = qb[(size_t)(c0 + 1) * HWn + pix];
      akf.x = kbp[(size_t)c0 * HWn + pix];
      akf.y = kbp[(size_t)(c0 + 1) * HWn + pix];
      esq += aqf.x * aqf.x + aqf.y * aqf.y;
      esk += akf.x * akf.x + akf.y * akf.y;
      // B fragments from LDS: 4 channels x 16 anchors, two anchor halves
      bq0.x = sAncQ[l15 * CC + c0];           bq0.y = sAncQ[l15 * CC + c0 + 1];
      bq1.x = sAncQ[(16 + l15) * CC + c0];    bq1.y = sAncQ[(16 + l15) * CC + c0 + 1];
      bk0.x = sAncK[l15 * CC + c0];           bk0.y = sAncK[l15 * CC + c0 + 1];
      bk1.x = sAncK[(16 + l15) * CC + c0];    bk1.y = sAncK[(16 + l15) * CC + c0 + 1];

      cq0 = __builtin_amdgcn_wmma_f32_16x16x4_f32(false, aqf, false, bq0, (short)0, cq0, false, false);
      cq1 = __builtin_amdgcn_wmma_f32_16x16x4_f32(false, aqf, false, bq1, (short)0, cq1, false, false);
      ck0 = __builtin_amdgcn_wmma_f32_16x16x4_f32(false, akf, false, bk0, (short)0, ck0, false, false);
      ck1 = __builtin_amdgcn_wmma_f32_16x16x4_f32(false, akf, false, bk1, (short)0, ck1, false, false);
    }

    // Fold ||e||^2 across the two K-halves: lane L and L+16 both get pixel m0+L15.
    esq += __shfl_xor(esq, 16, 32);
    esk += __shfl_xor(esk, 16, 32);

    // C/D layout: VGPR j holds M=j (lanes 0-15) / M=j+8 (lanes 16-31), N=lane&15.
    #pragma unroll
    for (int j = 0; j < 8; ++j) {
      float eqlo = __shfl(esq, j, 32), eqhi = __shfl(esq, j + 8, 32);
      float eklo = __shfl(esk, j, 32), ekhi = __shfl(esk, j + 8, 32);
      float eq = hi ? eqhi : eqlo;
      float ek = hi ? ekhi : eklo;

      float d, pq, pk;
      d  = fmaxf(eq - 2.0f * cq0[j] + aq0, 0.0f); pq = __expf(-d * INV_TS);
      d  = fmaxf(ek - 2.0f * ck0[j] + ak0, 0.0f); pk = __expf(-d * INV_TS);
      num0 += pq * pk; dq0 += pq * pq; dk0 += pk * pk;

      d  = fmaxf(eq - 2.0f * cq1[j] + aq1, 0.0f); pq = __expf(-d * INV_TS);
      d  = fmaxf(ek - 2.0f * ck1[j] + ak1, 0.0f); pk = __expf(-d * INV_TS);
      num1 += pq * pk; dq1 += pq * pq; dk1 += pk * pk;
    }
  }

  // Anchor n lives in lanes n and n+16: fold, then lanes 0-15 carry the sums.
  num0 += __shfl_xor(num0, 16, 32); num1 += __shfl_xor(num1, 16, 32);
  dq0  += __shfl_xor(dq0, 16, 32);  dq1  += __shfl_xor(dq1, 16, 32);
  dk0  += __shfl_xor(dk0, 16, 32);  dk1  += __shfl_xor(dk1, 16, 32);

  if (lane < 16) {
    atomicAdd(&sNum[l15], num0);      atomicAdd(&sNum[16 + l15], num1);
    atomicAdd(&sDq[l15], dq0);        atomicAdd(&sDq[16 + l15], dq1);
    atomicAdd(&sDk[l15], dk0);        atomicAdd(&sDk[16 + l15], dk1);
  }
  __syncthreads();

  if (tid < APAD) {
    atomicAdd(&gNum[b * APAD + tid], sNum[tid]);
    atomicAdd(&gDq[b * APAD + tid], sDq[tid]);
    atomicAdd(&gDk[b * APAD + tid], sDk[tid]);
  }
}

// ---------------------------------------------------------------------------
// Kernel 3: finalize — Dice score per (b, a<30), summed into scalar loss.
// ---------------------------------------------------------------------------
__global__ __launch_bounds__(128) void spoco_finalize(
    const float* __restrict__ gNum, const float* __restrict__ gDq,
    const float* __restrict__ gDk, float* __restrict__ out) {
  __shared__ float red[128];
  int t = threadIdx.x;
  float v = 0.0f;
  if (t < BB * AA) {
    int b = t / AA, a = t % AA;
    float n   = gNum[b * APAD + a];
    float den = gDq[b * APAD + a] + gDk[b * APAD + a];
    v = 1.0f - 2.0f * n / fmaxf(den, EPSF);
  }
  red[t] = v;
  __syncthreads();
  #pragma unroll
  for (int s = 64; s > 0; s >>= 1) {
    if (t < s) red[t] += red[t + s];
    __syncthreads();
  }
  if (t == 0) out[0] = red[0];
}

// ---------------------------------------------------------------------------
// Launch
// ---------------------------------------------------------------------------
extern "C" void kernel_launch(void* const* d_in, const int* in_sizes, int n_in,
                              void* d_out, int out_size, void* d_ws, size_t ws_size,
                              hipStream_t stream) {
  const float* emb_q = (const float*)d_in[0];
  const float* emb_k = (const float*)d_in[1];
  const int* anchor_idx = (const int*)d_in[2];
  float* out = (float*)d_out;

  float* ws   = (float*)d_ws;
  float* gNum = ws;            // [B*32]
  float* gDq  = ws + 128;      // [B*32]
  float* gDk  = ws + 256;      // [B*32]
  float* asqQ = ws + 384;      // [B*32]
  float* asqK = ws + 512;      // [B*32]
  float* ancQ = ws + 640;             // [B*32*64]
  float* ancK = ws + 640 + BB * APAD * CC;

  hipMemsetAsync(ws, 0, 384 * sizeof(float), stream);

  spoco_gather_anchors<<<256, 32, 0, stream>>>(emb_q, emb_k, anchor_idx,
                                               ancQ, ancK, asqQ, asqK);
  spoco_main<<<NBLOCKS, 256, 0, stream>>>(emb_q, emb_k, ancQ, ancK,
                                          asqQ, asqK, gNum, gDq, gDk);
  spoco_finalize<<<1, 128, 0, stream>>>(gNum, gDq, gDk, out);
}